// WindowAttention_46308337385637
// MI455X (gfx1250) — compile-verified
//
#include <hip/hip_runtime.h>
#include <hip/hip_bf16.h>
#include <math.h>

typedef __attribute__((ext_vector_type(2))) float v2f;
typedef __attribute__((ext_vector_type(8))) float v8f;

#define CIN   192
#define NTOT  13824      // 24*24*24
#define HEADS 6
#define BATCH 8
#define OC3   576

__device__ __forceinline__ v8f wmma4(v2f a, v2f b, v8f c) {
    // D = A(16x4 f32) * B(4x16 f32) + C(16x16 f32), full fp32 precision
    return __builtin_amdgcn_wmma_f32_16x16x4_f32(
        /*neg_a=*/false, a, /*neg_b=*/false, b,
        /*c_mod=*/(short)0, c, /*reuse_a=*/false, /*reuse_b=*/false);
}

// ---------------------------------------------------------------------------
// 1x1x1 conv == GEMM: Y[b,o,n] = sum_c W[o,c]*X[b,c,n] + bias[o]
// block: 256 thr (8 waves), tile 32(o) x 64(n); wave = 2(o) x 4(n) of 16x16
// X tile staged in LDS (stride 72 -> lane-groups 0-15/16-31 hit disjoint banks)
// W read directly from global (<=442KB, L2 resident), coalesced float2/lane.
// ---------------------------------------------------------------------------
__global__ __launch_bounds__(256) void conv1x1_gemm(
    const float* __restrict__ W, const float* __restrict__ X,
    const float* __restrict__ bias, float* __restrict__ Y, int OC)
{
    constexpr int XS = 72;
    __shared__ float xs[CIN * XS];          // 55296 bytes

    const int tid = threadIdx.x;
    const int b  = blockIdx.z;
    const int o0 = blockIdx.y * 32;
    const int n0 = blockIdx.x * 64;
    const float* Xb = X + (size_t)b * CIN * NTOT;

    // cooperative load of X tile (192 x 64), coalesced along n
    for (int idx = tid; idx < CIN * 64; idx += 256) {
        int c = idx >> 6, j = idx & 63;
        xs[c * XS + j] = Xb[(size_t)c * NTOT + n0 + j];
    }
    __syncthreads();

    const int wave = tid >> 5, lane = tid & 31;
    const int wo = wave >> 2;               // o half (0..1)
    const int wn = wave & 3;                // n quarter (0..3)
    const int m  = lane & 15, hi = lane >> 4;

    v8f acc = (v8f)0.0f;
    const float* wrow = W + (size_t)(o0 + wo * 16 + m) * CIN;
    const float* xcol = &xs[wn * 16 + m];

#pragma unroll 8
    for (int kk = 0; kk < CIN; kk += 4) {
        const int k2 = kk + 2 * hi;
        const float2 wv = *(const float2*)(wrow + k2);      // 8B aligned
        v2f a; a.x = wv.x; a.y = wv.y;
        v2f bf; bf.x = xcol[k2 * XS]; bf.y = xcol[(k2 + 1) * XS];
        acc = wmma4(a, bf, acc);
    }

    const int n = n0 + wn * 16 + m;
    float* Yb = Y + (size_t)b * OC * NTOT;
#pragma unroll
    for (int r = 0; r < 8; ++r) {
        const int om = o0 + wo * 16 + r + 8 * hi;           // D-layout row
        Yb[(size_t)om * NTOT + n] = acc[r] + bias[om];
    }
}

// ---------------------------------------------------------------------------
// XCA attention per (b,h). qkvbuf is the contiguous (B,576,N) conv output;
// the faithful-torch reshape means:
//   q[b,h,c,n] = qkvbuf[b, (n/24)*13824 + (n%24)*576 + h*32 + c]         (+0)
//   k: +192, v: +384.
// L2-normalization factors out: attn_raw[c,d] = S[c,d]*temp/(|q_c||k_d|).
// Phase1: 8 waves each accumulate 4 partial 16x16 S tiles + norm partials.
// Phase2: out[b, c*6+h, n] = attn @ v via WMMA, attn frags in registers.
// ---------------------------------------------------------------------------
__global__ __launch_bounds__(256) void xca_attn(
    const float* __restrict__ qkvbuf, const float* __restrict__ temperature,
    float* __restrict__ outbuf)
{
    constexpr int NW = 8;                   // waves per block
    __shared__ float red[NW * 1024];        // 32768 B : S partials
    __shared__ float nred[NW * 128];        // 4096 B  : norm partials
    __shared__ float attn[32 * 33];         // 4224 B
    __shared__ float nqk[64];

    const int bh = blockIdx.x;
    const int b = bh / HEADS, h = bh % HEADS;
    const float* Qb = qkvbuf + (size_t)b * OC3 * NTOT;

    const int tid = threadIdx.x, wave = tid >> 5, lane = tid & 31;
    const int m = lane & 15, hi = lane >> 4;
    const int hq = h * 32;

    // ---------------- Phase 1: S = q*k^T, row norms ----------------
    v8f Sll = (v8f)0.0f, Slh = (v8f)0.0f, Shl = (v8f)0.0f, Shh = (v8f)0.0f;
    float nql = 0.f, nqh = 0.f, nkl = 0.f, nkh = 0.f;

    const int nspan = NTOT / NW;            // 1728
    const int nstart = wave * nspan;
    for (int nb = nstart; nb < nstart + nspan; nb += 4) {
        const int na = nb + 2 * hi;         // this lane-group's K columns
        const int nb1 = na + 1;
        const int g0 = (na / 24) * 13824 + (na % 24) * 576 + hq;
        const int g1 = (nb1 / 24) * 13824 + (nb1 % 24) * 576 + hq;
        // lanes 0-15 read 16 consecutive floats (c contiguous) -> coalesced
        const float ql0 = Qb[g0 + m],       ql1 = Qb[g1 + m];
        const float qh0 = Qb[g0 + 16 + m],  qh1 = Qb[g1 + 16 + m];
        const float kl0 = Qb[g0 + 192 + m], kl1 = Qb[g1 + 192 + m];
        const float kh0 = Qb[g0 + 208 + m], kh1 = Qb[g1 + 208 + m];
        nql += ql0 * ql0 + ql1 * ql1;  nqh += qh0 * qh0 + qh1 * qh1;
        nkl += kl0 * kl0 + kl1 * kl1;  nkh += kh0 * kh0 + kh1 * kh1;
        v2f aql; aql.x = ql0; aql.y = ql1;
        v2f aqh; aqh.x = qh0; aqh.y = qh1;
        v2f bkl; bkl.x = kl0; bkl.y = kl1;
        v2f bkh; bkh.x = kh0; bkh.y = kh1;
        Sll = wmma4(aql, bkl, Sll);  Slh = wmma4(aql, bkh, Slh);
        Shl = wmma4(aqh, bkl, Shl);  Shh = wmma4(aqh, bkh, Shh);
    }

    {   // dump per-wave partials
        float* rw = &red[wave * 1024];
#pragma unroll
        for (int r = 0; r < 8; ++r) {
            rw[0 * 256 + lane * 8 + r] = Sll[r];
            rw[1 * 256 + lane * 8 + r] = Slh[r];
            rw[2 * 256 + lane * 8 + r] = Shl[r];
            rw[3 * 256 + lane * 8 + r] = Shh[r];
        }
        float* nw = &nred[wave * 128];
        nw[lane] = nql; nw[32 + lane] = nqh; nw[64 + lane] = nkl; nw[96 + lane] = nkh;
    }
    __syncthreads();

    // reduce S across waves; remap D-layout (tile,lane,r) -> (c,d)
    for (int v = tid; v < 1024; v += 256) {
        float s = 0.f;
#pragma unroll
        for (int w = 0; w < NW; ++w) s += red[w * 1024 + v];
        const int tile = v >> 8, li = (v >> 3) & 31, r = v & 7;
        const int c = (tile >> 1) * 16 + r + 8 * (li >> 4);
        const int d = (tile & 1) * 16 + (li & 15);
        attn[c * 33 + d] = s;
    }
    if (tid < 64) {                         // reduce norms: nq[0..31], nk[0..31]
        const int which = tid >> 5, c = tid & 31;
        const int quad = which * 2 + (c >> 4), cl = c & 15;
        float s = 0.f;
#pragma unroll
        for (int w = 0; w < NW; ++w)
            s += nred[w * 128 + quad * 32 + cl] + nred[w * 128 + quad * 32 + cl + 16];
        nqk[tid] = sqrtf(s);
    }
    __syncthreads();

    // ---------------- softmax rows (exact, fp32) ----------------
    if (tid < 32) {
        const int c = tid;
        const float temp = temperature[h];
        const float qn = fmaxf(nqk[c], 1e-12f);
        float vals[32], mx = -INFINITY;
#pragma unroll
        for (int d = 0; d < 32; ++d) {
            const float kn = fmaxf(nqk[32 + d], 1e-12f);
            const float vv = attn[c * 33 + d] / (qn * kn) * temp;
            vals[d] = vv; mx = fmaxf(mx, vv);
        }
        float sum = 0.f;
#pragma unroll
        for (int d = 0; d < 32; ++d) { vals[d] = expf(vals[d] - mx); sum += vals[d]; }
        const float inv = 1.f / sum;
#pragma unroll
        for (int d = 0; d < 32; ++d) attn[c * 33 + d] = vals[d] * inv;
    }
    __syncthreads();

    // ---------------- Phase 2: out = attn @ v ----------------
    float a0[2][8], a1[2][8];               // attn A-frags in registers
#pragma unroll
    for (int ch = 0; ch < 2; ++ch)
#pragma unroll
        for (int ks = 0; ks < 8; ++ks) {
            const int kd = ks * 4 + 2 * hi;
            a0[ch][ks] = attn[(ch * 16 + m) * 33 + kd];
            a1[ch][ks] = attn[(ch * 16 + m) * 33 + kd + 1];
        }

    float* Ob = outbuf + (size_t)b * CIN * NTOT;
    for (int chunk = wave; chunk < NTOT / 16; chunk += NW) {
        const int n2 = chunk * 16 + m;
        const int gv = (n2 / 24) * 13824 + (n2 % 24) * 576 + 384 + hq; // even
        v8f olo = (v8f)0.0f, ohi = (v8f)0.0f;
#pragma unroll
        for (int ks = 0; ks < 8; ++ks) {
            const int dd = ks * 4 + 2 * hi;
            const float2 vv = *(const float2*)(Qb + gv + dd);  // 8B aligned
            v2f bf; bf.x = vv.x; bf.y = vv.y;
            v2f alo; alo.x = a0[0][ks]; alo.y = a1[0][ks];
            v2f ahi; ahi.x = a0[1][ks]; ahi.y = a1[1][ks];
            olo = wmma4(alo, bf, olo);
            ohi = wmma4(ahi, bf, ohi);
        }
        const int ncol = chunk * 16 + m;
#pragma unroll
        for (int r = 0; r < 8; ++r) {
            const int cl = r + 8 * hi;
            Ob[(size_t)(cl * 6 + h) * NTOT + ncol]        = olo[r]; // ch = c*heads+h
            Ob[(size_t)((16 + cl) * 6 + h) * NTOT + ncol] = ohi[r];
        }
    }
}

// ---------------------------------------------------------------------------
extern "C" void kernel_launch(void* const* d_in, const int* in_sizes, int n_in,
                              void* d_out, int out_size, void* d_ws, size_t ws_size,
                              hipStream_t stream) {
    const float* x      = (const float*)d_in[0];
    const float* qkv_w  = (const float*)d_in[1];
    const float* qkv_b  = (const float*)d_in[2];
    const float* temp   = (const float*)d_in[3];
    const float* proj_w = (const float*)d_in[4];
    const float* proj_b = (const float*)d_in[5];
    float* out = (float*)d_out;

    float* qkvbuf = (float*)d_ws;                               // (B,576,N) f32
    float* obuf   = qkvbuf + (size_t)BATCH * OC3 * NTOT;        // (B,192,N) f32

    // 1) qkv projection: (576x192) @ (192xN) per batch
    dim3 g1(NTOT / 64, OC3 / 32, BATCH);
    conv1x1_gemm<<<g1, 256, 0, stream>>>(qkv_w, x, qkv_b, qkvbuf, OC3);

    // 2) channel cross-covariance attention per (b,h)
    xca_attn<<<dim3(BATCH * HEADS), 256, 0, stream>>>(qkvbuf, temp, obuf);

    // 3) output projection: (192x192) @ (192xN) per batch
    dim3 g3(NTOT / 64, CIN / 32, BATCH);
    conv1x1_gemm<<<g3, 256, 0, stream>>>(proj_w, obuf, proj_b, out, CIN);
}